// TransformerConvStack_2319282340277
// MI455X (gfx1250) — compile-verified
//
#include <hip/hip_runtime.h>
#include <hip/hip_bf16.h>

// ---------------------------------------------------------------------------
// TransformerConv x3 stack for MI455X (gfx1250, wave32).
//  - Node GEMMs (q/k/v/skip, final out) : V_WMMA_F32_16X16X4_F32
//  - Edge projection fused into edge kernels; We staged in LDS via the
//    Tensor Data Mover (tensor_load_to_lds + s_wait_tensorcnt)
//  - Softmax normalization folded into aggregation (single scatter pass)
//  - Lane owns 4 consecutive channels -> b128 LDS/global accesses
// ---------------------------------------------------------------------------

typedef float v2f __attribute__((ext_vector_type(2)));
typedef float v4f __attribute__((ext_vector_type(4)));
typedef float v8f __attribute__((ext_vector_type(8)));
typedef unsigned int u32x4 __attribute__((ext_vector_type(4)));
typedef int i32x4 __attribute__((ext_vector_type(4)));
typedef int i32x8 __attribute__((ext_vector_type(8)));

#define HC   128     // H*C
#define HDS  4       // heads
#define CDIM 32      // channels per head
#define NN   25000
#define EE   400000
#define WE_ELEMS (64 * HC)   // 8192 floats = 32 KB

#if defined(__has_builtin)
#  if __has_builtin(__builtin_amdgcn_tensor_load_to_lds) && __has_builtin(__builtin_amdgcn_s_wait_tensorcnt)
#    define USE_TDM 1
#  else
#    define USE_TDM 0
#  endif
#else
#  define USE_TDM 0
#endif

// ---------------------------------------------------------------------------
// Stage We (64x128 fp32, contiguous 32 KB) into LDS.
// TDM path: wave 0 issues one tensor_load_to_lds with a 1-D D# descriptor
// (count=1, data_size=4B, tensor_dim0=tile_dim0=8192, stride=8192), waits on
// TENSORcnt, then the workgroup barrier publishes LDS to all waves.
// This toolchain's builtin is the 6-arg form:
//   (u32x4 g0, i32x8 g1, i32x4 g2, i32x4 g3, i32x8 g4, i32 cpol)
// ---------------------------------------------------------------------------
__device__ __forceinline__ void stage_We(const float* __restrict__ We,
                                         float* __restrict__ sWe) {
#if USE_TDM
  if (threadIdx.x < 32) {                       // one wave issues the DMA
    unsigned long long ga = (unsigned long long)(size_t)We;
    unsigned lds = (unsigned)(size_t)(void*)sWe;
    u32x4 g0;
    g0[0] = 1u;                                  // count=1, user descriptor
    g0[1] = lds;                                 // lds_addr [63:32]
    g0[2] = (unsigned)(ga & 0xFFFFFFFFu);        // global_addr lo
    g0[3] = (unsigned)((ga >> 32) & 0x01FFFFFFu) // global_addr hi (bits 56:32)
            | (2u << 30);                        // type=2 ("image")
    i32x8 g1;
    g1[0] = (int)(2u << 16);                     // workgroup_mask=0, data_size=4B
    g1[1] = (int)((unsigned)WE_ELEMS << 16);     // tensor_dim0 lo16 in w1[31:16]
    g1[2] = (int)(1u << 16);                     // tensor_dim0 hi16=0 ; tensor_dim1=1
    g1[3] = (int)((unsigned)WE_ELEMS << 16);     // tensor_dim1 hi=0 ; tile_dim0=8192
    g1[4] = 1;                                   // tile_dim1=1, tile_dim2=0
    g1[5] = WE_ELEMS;                            // tensor_dim0_stride lo32
    g1[6] = 0;
    g1[7] = 0;
    i32x4 g2 = {0, 0, 0, 0};                     // unused (<=2D tensor)
    i32x4 g3 = {0, 0, 0, 0};
    i32x8 g4 = {0, 0, 0, 0, 0, 0, 0, 0};         // unused trailing group
    __builtin_amdgcn_tensor_load_to_lds(g0, g1, g2, g3, g4, 0);
    __builtin_amdgcn_s_wait_tensorcnt(0);
  }
  __syncthreads();
#else
  for (int i = threadIdx.x; i < WE_ELEMS; i += blockDim.x) sWe[i] = We[i];
  __syncthreads();
#endif
}

// ---------------------------------------------------------------------------
// GEMM: Y(M x NOUT) = [X1 | X2](M x (K1+K2)) @ W((K1+K2) x NOUT) + bias
// One wave = one 16x16 tile via V_WMMA_F32_16X16X4_F32; block = 4 waves.
// NOUT is a template constant so the unrolled B loads fold into constant
// load-immediate offsets (no per-iteration 64-bit address VALU).
// fp32 fragment layouts (ISA 7.12.2):
//   A: lane m=lane&15, VGPR v holds K = k + v + 2*(lane>>4)  -> one v2f load
//   B: lane n=lane&15, VGPR v holds K row = k + v + 2*(lane>>4)
//   C: VGPR r holds row M = r + 8*(lane>>4), col N = lane&15
// ---------------------------------------------------------------------------
template <int NOUT>
__global__ void gemm_bias_wmma(const float* __restrict__ X1, int K1,
                               const float* __restrict__ X2, int K2,
                               const float* __restrict__ W,
                               const float* __restrict__ bias,
                               float* __restrict__ Y, int M) {
  const int wave = threadIdx.x >> 5;
  const int lane = threadIdx.x & 31;
  const int ct   = blockIdx.y * 4 + wave;        // 16-wide column tile index
  if (ct * 16 >= NOUT) return;                   // wave-uniform: EXEC stays all-1s
  const int half = lane >> 4;                    // 0 | 1
  const int lm   = lane & 15;
  const int row0 = blockIdx.x * 16;
  int arow = row0 + lm; if (arow >= M) arow = M - 1;   // clamp (stores are guarded)
  const int n = ct * 16 + lm;

  const float* __restrict__ A1 = X1 + (size_t)arow * K1 + 2 * half;
  const float* __restrict__ Wn = W + (size_t)(2 * half) * NOUT + n;

  v8f acc = {};
  for (int k = 0; k < K1; k += 4) {
    const v2f a = *(const v2f*)(A1 + k);
    v2f b;
    b[0] = Wn[(size_t)k * NOUT];
    b[1] = Wn[(size_t)k * NOUT + NOUT];
    acc = __builtin_amdgcn_wmma_f32_16x16x4_f32(false, a, false, b,
                                                (short)0, acc, false, false);
  }
  if (K2 > 0) {
    const float* __restrict__ A2 = X2 + (size_t)arow * K2 + 2 * half;
    const float* __restrict__ W2 = Wn + (size_t)K1 * NOUT;
    for (int k = 0; k < K2; k += 4) {
      const v2f a = *(const v2f*)(A2 + k);
      v2f b;
      b[0] = W2[(size_t)k * NOUT];
      b[1] = W2[(size_t)k * NOUT + NOUT];
      acc = __builtin_amdgcn_wmma_f32_16x16x4_f32(false, a, false, b,
                                                  (short)0, acc, false, false);
    }
  }

  const float bn = bias ? bias[n] : 0.0f;
#pragma unroll
  for (int r = 0; r < 8; ++r) {
    const int rr = row0 + r + 8 * half;
    if (rr < M) Y[(size_t)rr * NOUT + n] = acc[r] + bn;
  }
}

// ---------------------------------------------------------------------------
// Per-layer scratch init: agg=0, amax=-inf, denom=0
// ---------------------------------------------------------------------------
__global__ void init_buffers(float* __restrict__ agg, float* __restrict__ amax,
                             float* __restrict__ denom, int N) {
  const int idx = blockIdx.x * blockDim.x + threadIdx.x;
  if (idx < N * HC) agg[idx] = 0.0f;
  if (idx < N * HDS) { amax[idx] = -3.0e38f; denom[idx] = 0.0f; }
}

// ---------------------------------------------------------------------------
// Pass 1 over edges (wave-per-edge, grid-stride; lane owns channels c0..c0+3):
//   e      = ea[edge] @ We          (We LDS-resident, b128 reads)
//   alpha  = <q[dst], k[src]+e> / sqrt(C)  per head (8-lane shfl_xor reduce)
//   amax[dst] = segment-max(alpha)  (fp32 atomicMax)
// ---------------------------------------------------------------------------
__global__ void edge_alpha(const float* __restrict__ q, const float* __restrict__ kmat,
                           const float* __restrict__ ea, const float* __restrict__ We,
                           const long long* __restrict__ src,
                           const long long* __restrict__ dst,
                           float* __restrict__ alpha, float* __restrict__ amax,
                           int E) {
  __shared__ __align__(16) float sWe[WE_ELEMS];
  stage_We(We, sWe);

  const int wave  = threadIdx.x >> 5;
  const int lane  = threadIdx.x & 31;
  const int c0    = lane * 4;                    // 4 consecutive channels
  const int h     = lane >> 3;                   // head of this lane group
  const int wstep = gridDim.x * (blockDim.x >> 5);

  for (int edge = blockIdx.x * (blockDim.x >> 5) + wave; edge < E; edge += wstep) {
    const int s = (int)src[edge];
    const int d = (int)dst[edge];

    // edge projection (4 channels per lane, b128 LDS reads)
    v4f e4 = {0.f, 0.f, 0.f, 0.f};
    for (int j = 0; j < 64; ++j) {
      const float eaj = ea[(size_t)edge * 64 + j];   // wave-uniform broadcast
      const v4f w = *(const v4f*)(sWe + j * HC + c0);
      e4 += w * eaj;
    }

    const v4f k4 = *(const v4f*)(kmat + (size_t)s * HC + c0);
    const v4f q4 = *(const v4f*)(q    + (size_t)d * HC + c0);
    float p = q4[0] * (k4[0] + e4[0]) + q4[1] * (k4[1] + e4[1])
            + q4[2] * (k4[2] + e4[2]) + q4[3] * (k4[3] + e4[3]);
#pragma unroll
    for (int off = 4; off; off >>= 1) p += __shfl_xor(p, off, 32);  // 8-lane group

    if ((lane & 7) == 0) {
      const float a = p * 0.17677669529663687f;     // 1/sqrt(32)
      alpha[(size_t)edge * HDS + h] = a;
      atomicMax(&amax[(size_t)d * HDS + h], a);
    }
  }
}

// ---------------------------------------------------------------------------
// Pass 2 over edges:
//   ex   = exp(alpha - amax[dst])   -> atomicAdd denom[dst]
//   agg[dst] += ex * (v[src] + e)   (unnormalized; divided in finalize)
// ---------------------------------------------------------------------------
__global__ void edge_msg(const float* __restrict__ vmat, const float* __restrict__ ea,
                         const float* __restrict__ We,
                         const long long* __restrict__ src,
                         const long long* __restrict__ dst,
                         const float* __restrict__ alpha, const float* __restrict__ amax,
                         float* __restrict__ denom, float* __restrict__ agg, int E) {
  __shared__ __align__(16) float sWe[WE_ELEMS];
  stage_We(We, sWe);

  const int wave  = threadIdx.x >> 5;
  const int lane  = threadIdx.x & 31;
  const int c0    = lane * 4;
  const int h     = lane >> 3;
  const int wstep = gridDim.x * (blockDim.x >> 5);

  for (int edge = blockIdx.x * (blockDim.x >> 5) + wave; edge < E; edge += wstep) {
    const int s = (int)src[edge];
    const int d = (int)dst[edge];

    float exl = 0.0f;
    if (lane < HDS) {
      exl = __expf(alpha[(size_t)edge * HDS + lane] - amax[(size_t)d * HDS + lane]);
      atomicAdd(&denom[(size_t)d * HDS + lane], exl);
    }
    const float exh = __shfl(exl, h, 32);         // broadcast this lane's head weight

    v4f e4 = {0.f, 0.f, 0.f, 0.f};
    for (int j = 0; j < 64; ++j) {
      const float eaj = ea[(size_t)edge * 64 + j];
      const v4f w = *(const v4f*)(sWe + j * HC + c0);
      e4 += w * eaj;
    }

    const v4f v4 = *(const v4f*)(vmat + (size_t)s * HC + c0);
    float* ag = agg + (size_t)d * HC + c0;
#pragma unroll
    for (int t = 0; t < 4; ++t) atomicAdd(ag + t, (v4[t] + e4[t]) * exh);
  }
}

// ---------------------------------------------------------------------------
// Finalize: h_out = agg/(denom+eps) + (h_in @ Ws + bs)   [skip precomputed]
// ---------------------------------------------------------------------------
__global__ void node_finalize(const float* __restrict__ agg, const float* __restrict__ denom,
                              const float* __restrict__ skip, float* __restrict__ hout,
                              int N) {
  const int idx = blockIdx.x * blockDim.x + threadIdx.x;
  if (idx >= N * HC) return;
  const int n = idx >> 7;
  const int h = (idx & (HC - 1)) >> 5;
  hout[idx] = agg[idx] / (denom[n * HDS + h] + 1e-16f) + skip[idx];
}

// ---------------------------------------------------------------------------
// Host orchestration
// ---------------------------------------------------------------------------
extern "C" void kernel_launch(void* const* d_in, const int* in_sizes, int n_in,
                              void* d_out, int out_size, void* d_ws, size_t ws_size,
                              hipStream_t stream) {
  const int N = NN, E = EE;
  const float*     x   = (const float*)d_in[0];
  const long long* ei  = (const long long*)d_in[1];     // (2,E) int64
  const float*     ea  = (const float*)d_in[2];
  const long long* src = ei;
  const long long* dst = ei + E;

  // workspace carve-up (floats)
  float* ws    = (float*)d_ws;
  float* q     = ws;               // N*128
  float* k     = q     + (size_t)N * HC;
  float* v     = k     + (size_t)N * HC;
  float* skip  = v     + (size_t)N * HC;
  float* h0    = skip  + (size_t)N * HC;
  float* h1    = h0    + (size_t)N * HC;
  float* agg   = h1    + (size_t)N * HC;
  float* alpha = agg   + (size_t)N * HC;                // E*4
  float* amax  = alpha + (size_t)E * HDS;               // N*4
  float* denom = amax  + (size_t)N * HDS;               // N*4

  const dim3 gemmGrid((N + 15) / 16, 2);                // 128 output cols
  const dim3 gemmBlock(128);
  const int  elemBlocks = (N * HC + 255) / 256;
  const int  edgeBlocks = 12500;                        // 8 waves/block, 4 edges/wave

  auto run_layer = [&](const float* X1, int K1, const float* X2, int K2,
                       int p, float* hout) {
    const float* Wq = (const float*)d_in[p + 0];
    const float* bq = (const float*)d_in[p + 1];
    const float* Wk = (const float*)d_in[p + 2];
    const float* bk = (const float*)d_in[p + 3];
    const float* Wv = (const float*)d_in[p + 4];
    const float* bv = (const float*)d_in[p + 5];
    const float* We = (const float*)d_in[p + 6];
    const float* Ws = (const float*)d_in[p + 7];
    const float* bs = (const float*)d_in[p + 8];

    init_buffers<<<elemBlocks, 256, 0, stream>>>(agg, amax, denom, N);
    gemm_bias_wmma<HC><<<gemmGrid, gemmBlock, 0, stream>>>(X1, K1, X2, K2, Wq, bq, q,    N);
    gemm_bias_wmma<HC><<<gemmGrid, gemmBlock, 0, stream>>>(X1, K1, X2, K2, Wk, bk, k,    N);
    gemm_bias_wmma<HC><<<gemmGrid, gemmBlock, 0, stream>>>(X1, K1, X2, K2, Wv, bv, v,    N);
    gemm_bias_wmma<HC><<<gemmGrid, gemmBlock, 0, stream>>>(X1, K1, X2, K2, Ws, bs, skip, N);
    edge_alpha<<<edgeBlocks, 256, 0, stream>>>(q, k, ea, We, src, dst, alpha, amax, E);
    edge_msg  <<<edgeBlocks, 256, 0, stream>>>(v, ea, We, src, dst, alpha, amax, denom, agg, E);
    node_finalize<<<elemBlocks, 256, 0, stream>>>(agg, denom, skip, hout, N);
  };

  run_layer(x,  256, nullptr, 0,   3,  h0);   // layer 0: input = x (256)
  run_layer(h0, 128, x,       256, 12, h1);   // layer 1: input = [h0 | x] (384)
  run_layer(h1, 128, x,       256, 21, h0);   // layer 2: input = [h1 | x] (384)

  // final projection: out = h @ Wout + bout   (N x 32)
  const float* Wout = (const float*)d_in[30];
  const float* bout = (const float*)d_in[31];
  gemm_bias_wmma<CDIM><<<dim3((N + 15) / 16, 1), gemmBlock, 0, stream>>>(
      h0, HC, nullptr, 0, Wout, bout, (float*)d_out, N);
}